// SuppAlignLayer_50233937494365
// MI455X (gfx1250) — compile-verified
//
#include <hip/hip_runtime.h>
#include <stdint.h>

// ROI-Align over a 4-level FPN pyramid (see reference): out[l][r][c][7][7].
// Block = (roi, level), 128 threads (4 wave32s). Per-ROI bilinear weights are
// computed once into LDS (validity folded into zeroed weights). The feature
// sub-tile (x, y, channel-chunk) is staged into LDS with the CDNA5 Tensor
// Data Mover (tensor_load_to_lds), double-buffered on TENSORcnt. Outputs are
// written with non-temporal stores so the ~165MB feature pyramid stays
// resident in the 192MB L2 while the ~205MB of outputs stream through.

#define CCH   256          // channels
#define OUTS  7            // output size
#define GCH   2            // channels per TDM chunk
#define TDIM  62           // max tile side (level0: rw<=64 -> span<=62)
#define TILE_ELEMS (GCH * TDIM * TDIM)

typedef unsigned int u32;
typedef u32 u32x4 __attribute__((ext_vector_type(4)));
typedef int i32x4 __attribute__((ext_vector_type(4)));
typedef int i32x8 __attribute__((ext_vector_type(8)));

// Build a D# per CDNA5 ISA (CDNA5 ISA) and issue the DMA.
// 3-D tile: dim0 = Xt (x, stride 1), dim1 = Yt (y, stride W), dim2 = GCH
// (channels, stride H*W). Tile is fully in-bounds by construction.
__device__ __forceinline__ void tdm_load_tile(const float* gptr, u32 lds_byte,
                                              int Xt, int Yt, int W, int HW) {
  unsigned long long ga = (unsigned long long)(uintptr_t)gptr;

  u32x4 g0;
  g0[0] = 1u;                                            // count=1 (valid user D#)
  g0[1] = lds_byte;                                      // lds_addr
  g0[2] = (u32)ga;                                       // global_addr[31:0]
  g0[3] = ((u32)(ga >> 32) & 0x01FFFFFFu) | (2u << 30);  // ga[56:32] | type=2

  i32x8 g1;
  g1[0] = (int)(2u << 16);                               // data_size=4B, no mask/pad
  g1[1] = (int)(((u32)Xt & 0xFFFFu) << 16);              // tensor_dim0[15:0] @63:48
  g1[2] = (int)(((u32)Xt >> 16) | (((u32)Yt & 0xFFFFu) << 16)); // td0 hi | td1 lo
  g1[3] = (int)(((u32)Yt >> 16) | ((u32)Xt << 16));      // td1 hi | tile_dim0
  g1[4] = (int)((u32)Yt | ((u32)GCH << 16));             // tile_dim1 | tile_dim2
  g1[5] = W;                                             // tensor_dim0_stride[31:0]
  g1[6] = (int)(((u32)HW & 0xFFFFu) << 16);              // stride0 hi=0 | stride1[15:0]
  g1[7] = (int)((u32)HW >> 16);                          // stride1[47:16]

  i32x4 g2;
  g2[0] = GCH;                                           // tensor_dim2
  g2[1] = 0;                                             // tensor_dim3 unused
  g2[2] = 0;                                             // tensor_dim2_stride (unused 3D)
  g2[3] = 0;                                             // tile_dim3 = 0 -> 3D tile

  i32x4 g3 = {0, 0, 0, 0};
  i32x8 g4 = {0, 0, 0, 0, 0, 0, 0, 0};                   // unused trailing group

  __builtin_amdgcn_tensor_load_to_lds(g0, g1, g2, g3, g4, 0);
}

__global__ __launch_bounds__(128)
void roi_align_fpn_kernel(const float* __restrict__ f0, const float* __restrict__ f1,
                          const float* __restrict__ f2, const float* __restrict__ f3,
                          const float* __restrict__ boxes, float* __restrict__ out,
                          int R, int N) {
  __shared__ int   sx0[14], sx1[14], sy0[14], sy1[14];
  __shared__ float shx[14], slx[14], shy[14], sly[14];
  __shared__ float tile[2][TILE_ELEMS];

  const int r   = blockIdx.x;
  const int lvl = blockIdx.y;
  const int tid = threadIdx.x;

  const int   H     = 200 >> lvl;            // 200,100,50,25
  const int   W     = 304 >> lvl;            // 304,152,76,38
  const float scale = 0.25f / (float)(1 << lvl);
  const float* feat = (lvl == 0) ? f0 : (lvl == 1) ? f1 : (lvl == 2) ? f2 : f3;
  const int b = r / N;

  // ---- per-ROI sample grid: 14 x-positions (tid 0..13), 14 y (tid 14..27) ----
  if (tid < 28) {
    const float bx1 = boxes[r * 4 + 0] * scale;
    const float by1 = boxes[r * 4 + 1] * scale;
    const float bx2 = boxes[r * 4 + 2] * scale;
    const float by2 = boxes[r * 4 + 3] * scale;
    const float rw = fmaxf(bx2 - bx1, 1.0f);
    const float rh = fmaxf(by2 - by1, 1.0f);
    const int  isx  = (tid < 14) ? 1 : 0;
    const int  k    = isx ? tid : (tid - 14);
    const float beg = isx ? bx1 : by1;
    const float len = isx ? rw : rh;
    const int  ilim = isx ? W : H;
    const float s = beg + len * (((float)k + 0.5f) / 14.0f);
    const bool  v = (s >= -1.0f) && (s <= (float)ilim);
    float c = fminf(fmaxf(s, 0.0f), (float)ilim - 1.0f);
    int i0 = (int)floorf(c);
    int i1 = i0 + 1; if (i1 > ilim - 1) i1 = ilim - 1;
    float l = c - (float)i0;
    float h = 1.0f - l;
    if (!v) { l = 0.0f; h = 0.0f; }                 // fold validity into weights
    if (isx) { sx0[k] = i0; sx1[k] = i1; slx[k] = l; shx[k] = h; }
    else     { sy0[k] = i0; sy1[k] = i1; sly[k] = l; shy[k] = h; }
  }
  __syncthreads();

  // Sample positions are monotone -> bounding tile from first/last corners.
  const int xmin = sx0[0], ymin = sy0[0];
  int Xt = sx1[13] - xmin + 1; if (Xt > TDIM) Xt = TDIM;
  int Yt = sy1[13] - ymin + 1; if (Yt > TDIM) Yt = TDIM;
  const int HW = H * W;

  const int nchunks = CCH / GCH;
  const size_t out_base = ((size_t)lvl * (size_t)R + (size_t)r) *
                          (size_t)(CCH * OUTS * OUTS);

  // Prime the pipeline: wave 0 DMAs channel chunk 0.
  if (tid < 32) {
    const float* g =
        feat + (size_t)(b * CCH) * HW + (size_t)ymin * W + xmin;
    tdm_load_tile(g, (u32)(uintptr_t)&tile[0][0], Xt, Yt, W, HW);
  }

  for (int k = 0; k < nchunks; ++k) {
    __syncthreads();  // all compute on buffer (k+1)&1 from iter k-1 finished
    if (tid < 32) {
      if (k + 1 < nchunks) {
        const int c0 = (k + 1) * GCH;
        const float* g =
            feat + (size_t)(b * CCH + c0) * HW + (size_t)ymin * W + xmin;
        tdm_load_tile(g, (u32)(uintptr_t)&tile[(k + 1) & 1][0], Xt, Yt, W, HW);
        __builtin_amdgcn_s_wait_tensorcnt(1);  // chunk k's DMA complete
      } else {
        __builtin_amdgcn_s_wait_tensorcnt(0);
      }
    }
    __syncthreads();  // tile k visible to all waves

    // 98 threads: one (channel, pixel) output each; 2x2 samples x 4 corners.
    if (tid < OUTS * OUTS * GCH) {
      const int c  = tid / (OUTS * OUTS);
      const int p  = tid % (OUTS * OUTS);
      const int py = p / OUTS, px = p % OUTS;
      const float* pl = &tile[k & 1][c * (Yt * Xt)];
      float acc = 0.0f;
#pragma unroll
      for (int di = 0; di < 2; ++di) {
        const int i   = 2 * py + di;
        const int iy0 = (sy0[i] - ymin) * Xt;
        const int iy1 = (sy1[i] - ymin) * Xt;
        const float wh = shy[i], wl = sly[i];
#pragma unroll
        for (int dj = 0; dj < 2; ++dj) {
          const int j   = 2 * px + dj;
          const int jx0 = sx0[j] - xmin;
          const int jx1 = sx1[j] - xmin;
          const float v0 = shx[j] * pl[iy0 + jx0] + slx[j] * pl[iy0 + jx1];
          const float v1 = shx[j] * pl[iy1 + jx0] + slx[j] * pl[iy1 + jx1];
          acc += wh * v0 + wl * v1;
        }
      }
      // Non-temporal: keep the (L2-resident) feature pyramid from being
      // evicted by the streamed 205MB output tensor.
      __builtin_nontemporal_store(
          0.25f * acc,
          &out[out_base + (size_t)(k * GCH + c) * (OUTS * OUTS) + p]);
    }
  }
}

extern "C" void kernel_launch(void* const* d_in, const int* in_sizes, int n_in,
                              void* d_out, int out_size, void* d_ws, size_t ws_size,
                              hipStream_t stream) {
  const float* f0    = (const float*)d_in[0];
  const float* f1    = (const float*)d_in[1];
  const float* f2    = (const float*)d_in[2];
  const float* f3    = (const float*)d_in[3];
  const float* boxes = (const float*)d_in[4];

  int B = in_sizes[0] / (CCH * 200 * 304);
  if (B < 1) B = 1;
  int N = in_sizes[4] / (4 * B);
  int R = B * N;

  dim3 grid((unsigned)R, 4, 1);
  dim3 block(128, 1, 1);
  roi_align_fpn_kernel<<<grid, block, 0, stream>>>(
      f0, f1, f2, f3, boxes, (float*)d_out, R, N);
}